// SpGraphAttentionLayer_16612933501032
// MI455X (gfx1250) — compile-verified
//
#include <hip/hip_runtime.h>
#include <math.h>

// GAT layer, refactored:
//   H = X @ [W1|W2]^T  (one dense 50000x128x256 GEMM via v_wmma_f32_16x16x4_f32)
//   s1 = H1@a, s2 = H2@a
//   per edge: ee = exp(-lrelu(s1[src]+s2[dst])); esum[src]+=ee; acc[src]+=ee*H2[dst]
//   out[n] = elu((esum[n]*H1[n] + acc[n]) / max(esum[n],eps))
// All hot buffers (H 51MB, acc 25.6MB) are L2-resident on MI455X (192MB L2).

typedef __attribute__((ext_vector_type(2))) float v2f;
typedef __attribute__((ext_vector_type(8))) float v8f;

#define N_NODES 50000
#define N_EDGES 800000
#define F 128
#define TWO_F 256
#define LRELU_ALPHA 0.2f
#define GAT_EPS 1e-12f

// workspace layout in floats
#define H_OFF    0ull
#define S1_OFF   ((unsigned long long)N_NODES * TWO_F)          // 12,800,000
#define S2_OFF   (S1_OFF + N_NODES)
#define ESUM_OFF (S2_OFF + N_NODES)
#define ACC_OFF  (ESUM_OFF + N_NODES)
// total: 12.8M + 150K + 6.4M floats ~= 77.4 MB

__global__ __launch_bounds__(256) void gat_zero(float* __restrict__ p, unsigned long long count) {
  unsigned long long i = (unsigned long long)blockIdx.x * 256ull + threadIdx.x;
  if (i < count) p[i] = 0.0f;
}

// Dense GEMM: H[n][o] = sum_i X[n][i] * B[i][o],
//   B[i][o] = W[o][i]        (o < 128)  -> column o of B is row o of W (contiguous in i)
//   B[i][o] = W[o-128][128+i] (o >= 128)
// One wave computes a 16x64 strip (4 WMMA accumulators), K = 128 in steps of 4.
__global__ __launch_bounds__(256) void gat_gemm_wmma(const float* __restrict__ X,
                                                     const float* __restrict__ W,
                                                     float* __restrict__ H) {
  const int wid  = (blockIdx.x * 256 + (int)threadIdx.x) >> 5;
  const int lane = threadIdx.x & 31;
  const int row_tile = wid >> 2;
  if (row_tile >= N_NODES / 16) return;           // uniform per wave: EXEC stays all-ones
  const int col0 = (wid & 3) * 64;
  const int row0 = row_tile * 16;

  const int m    = lane & 15;                      // A: row index / B: col index within tile
  const int koff = (lane >> 4) << 1;               // lanes 0-15: K=0,1  lanes 16-31: K=2,3

  const float* arow = X + (size_t)(row0 + m) * F + koff;
  const float* bcol[4];
#pragma unroll
  for (int c = 0; c < 4; ++c) {
    int o = col0 + c * 16 + m;                     // uniform branch: strips are 64-aligned
    bcol[c] = (o < F) ? (W + (size_t)o * TWO_F + koff)
                      : (W + (size_t)(o - F) * TWO_F + F + koff);
  }

  v8f acc0 = {}, acc1 = {}, acc2 = {}, acc3 = {};
#pragma unroll 4
  for (int k0 = 0; k0 < F; k0 += 4) {
    v2f A  = *(const v2f*)(arow + k0);
    v2f B0 = *(const v2f*)(bcol[0] + k0);
    v2f B1 = *(const v2f*)(bcol[1] + k0);
    v2f B2 = *(const v2f*)(bcol[2] + k0);
    v2f B3 = *(const v2f*)(bcol[3] + k0);
    acc0 = __builtin_amdgcn_wmma_f32_16x16x4_f32(false, A, false, B0, (short)0, acc0, false, false);
    acc1 = __builtin_amdgcn_wmma_f32_16x16x4_f32(false, A, false, B1, (short)0, acc1, false, false);
    acc2 = __builtin_amdgcn_wmma_f32_16x16x4_f32(false, A, false, B2, (short)0, acc2, false, false);
    acc3 = __builtin_amdgcn_wmma_f32_16x16x4_f32(false, A, false, B3, (short)0, acc3, false, false);
  }

  // C/D layout: VGPR j -> row (j + 8*(lane>=16)), col = lane%16
  const int rbase = row0 + ((lane >> 4) << 3);
#pragma unroll
  for (int j = 0; j < 8; ++j) {
    size_t r = (size_t)(rbase + j) * TWO_F + col0 + m;
    H[r +  0] = acc0[j];
    H[r + 16] = acc1[j];
    H[r + 32] = acc2[j];
    H[r + 48] = acc3[j];
  }
}

// s1[n] = H1[n] . a ; s2[n] = H2[n] . a   (one wave per node, float4 lanes + shuffle reduce)
__global__ __launch_bounds__(256) void gat_scores(const float* __restrict__ H,
                                                  const float* __restrict__ a,
                                                  float* __restrict__ s1,
                                                  float* __restrict__ s2) {
  const int wid  = (blockIdx.x * 256 + (int)threadIdx.x) >> 5;
  const int lane = threadIdx.x & 31;
  if (wid >= N_NODES) return;
  float4 av = ((const float4*)a)[lane];
  float4 x1 = ((const float4*)(H + (size_t)wid * TWO_F))[lane];
  float4 x2 = ((const float4*)(H + (size_t)wid * TWO_F + F))[lane];
  float p1 = x1.x * av.x + x1.y * av.y + x1.z * av.z + x1.w * av.w;
  float p2 = x2.x * av.x + x2.y * av.y + x2.z * av.z + x2.w * av.w;
#pragma unroll
  for (int off = 16; off > 0; off >>= 1) {
    p1 += __shfl_down(p1, off, 32);
    p2 += __shfl_down(p2, off, 32);
  }
  if (lane == 0) { s1[wid] = p1; s2[wid] = p2; }
}

// One wave per edge: ee scalar + 128-wide scatter into L2-resident accumulators.
__global__ __launch_bounds__(256) void gat_edges(const int* __restrict__ edge,
                                                 const float* __restrict__ H,
                                                 const float* __restrict__ s1,
                                                 const float* __restrict__ s2,
                                                 float* __restrict__ esum,
                                                 float* __restrict__ acc) {
  const int wid  = (blockIdx.x * 256 + (int)threadIdx.x) >> 5;
  const int lane = threadIdx.x & 31;
  if (wid >= N_EDGES) return;
  const int src = edge[wid];
  const int dst = edge[N_EDGES + wid];

  const float* h2 = H + (size_t)dst * TWO_F + F;
  __builtin_prefetch(h2 + lane * 4, 0, 0);         // global_prefetch of the H2 row

  const float sc = s1[src] + s2[dst];
  const float lr = sc > 0.0f ? sc : LRELU_ALPHA * sc;
  const float ee = __expf(-lr);

  if (lane == 0) atomicAdd(&esum[src], ee);
  float* o = acc + (size_t)src * F;
#pragma unroll
  for (int r = 0; r < 4; ++r) {
    const int f = lane + (r << 5);
    atomicAdd(&o[f], ee * h2[f]);
  }
}

// out[n][f] = elu( (esum[n]*H1[n][f] + acc[n][f]) / denom )
__global__ __launch_bounds__(256) void gat_finalize(const float* __restrict__ H,
                                                    const float* __restrict__ esum,
                                                    const float* __restrict__ acc,
                                                    float* __restrict__ out) {
  const unsigned long long idx = (unsigned long long)blockIdx.x * 256ull + threadIdx.x;
  if (idx >= (unsigned long long)N_NODES * F) return;
  const int n = (int)(idx >> 7);
  const int f = (int)(idx & 127);
  const float es = esum[n];
  const float denom = (es == 0.0f) ? GAT_EPS : es;
  const float v = (es * H[(size_t)n * TWO_F + f] + acc[idx]) / denom;
  out[idx] = v > 0.0f ? v : (__expf(v) - 1.0f);
}

extern "C" void kernel_launch(void* const* d_in, const int* in_sizes, int n_in,
                              void* d_out, int out_size, void* d_ws, size_t ws_size,
                              hipStream_t stream) {
  (void)in_sizes; (void)n_in; (void)out_size; (void)ws_size;
  const float* X    = (const float*)d_in[0];   // [50000,128]
  const float* W    = (const float*)d_in[1];   // [128,256]
  const float* a    = (const float*)d_in[2];   // [1,128]
  const int*   edge = (const int*)d_in[3];     // [2,800000]

  float* ws   = (float*)d_ws;
  float* H    = ws + H_OFF;
  float* s1   = ws + S1_OFF;
  float* s2   = ws + S2_OFF;
  float* esum = ws + ESUM_OFF;
  float* acc  = ws + ACC_OFF;
  float* out  = (float*)d_out;

  // zero esum + acc (contiguous)
  const unsigned long long zc = (unsigned long long)N_NODES * (F + 1);
  gat_zero<<<(unsigned)((zc + 255ull) / 256ull), 256, 0, stream>>>(esum, zc);

  // GEMM: 3125 row tiles x 4 col strips = 12500 waves, 8 waves/block
  gat_gemm_wmma<<<1563, 256, 0, stream>>>(X, W, H);

  // scores: 50000 waves
  gat_scores<<<6250, 256, 0, stream>>>(H, a, s1, s2);

  // edges: 800000 waves
  gat_edges<<<100000, 256, 0, stream>>>(edge, H, s1, s2, esum, acc);

  // finalize: 6.4M threads
  gat_finalize<<<(N_NODES * F + 255) / 256, 256, 0, stream>>>(H, esum, acc, out);
}